// GAE_23897198035293
// MI455X (gfx1250) — compile-verified
//
#include <hip/hip_runtime.h>
#include <hip/hip_bf16.h>

typedef __attribute__((ext_vector_type(2))) float v2f;
typedef __attribute__((ext_vector_type(8))) float v8f;
typedef __attribute__((ext_vector_type(4))) unsigned int v4u;
typedef __attribute__((ext_vector_type(8))) int v8i;
typedef __attribute__((ext_vector_type(4))) int v4i;

#define N_NODES 9746
#define N_USER  6040
#define N_ITEM  3706   // N_NODES - N_USER
#define HID     500
#define OUTC    75
#define NREL    5
#define NEDGE   200000
#define NBASIS  2
#define UPAD    6048   // N_USER padded to 16
#define IPAD    3712   // N_ITEM padded to 16
#define KPAD    80     // OUTC padded to multiple of 4 (WMMA K step)

// macro-tile for the fused score kernel
#define BU      32     // u rows per block
#define BI      64     // i rows per block
#define SA_F    (NREL * BU * KPAD)   // 12800 floats (51200 B) uQ slab in LDS
#define SB_F    (BI * KPAD)          //  5120 floats (20480 B) projI slab in LDS

// ---- workspace layout (float offsets) ----
#define OFF_W   ((size_t)0)                        // relu(cumsum(ord_basis)) [5][9746][500]
#define SZ_W    ((size_t)NREL * N_NODES * HID)
#define OFF_F   (OFF_W + SZ_W)                     // feats [9746][500]
#define SZ_F    ((size_t)N_NODES * HID)
#define OFF_PU  (OFF_F + SZ_F)                     // projU [6048][80] (zero padded)
#define SZ_PU   ((size_t)UPAD * KPAD)
#define OFF_PI  (OFF_PU + SZ_PU)                   // projI [3712][80] (zero padded)
#define SZ_PI   ((size_t)IPAD * KPAD)
#define OFF_Q   (OFF_PI + SZ_PI)                   // Q [5][75][75]
#define SZ_Q    ((size_t)NREL * OUTC * OUTC)
#define OFF_UQ  (OFF_Q + SZ_Q)                     // uQ [5][6048][80] (zero padded)
#define SZ_UQ   ((size_t)NREL * UPAD * KPAD)
#define ZERO_N  (SZ_F + SZ_PU + SZ_PI + SZ_Q + SZ_UQ)

// ---------------- LDS offset helper (as3 ptrtoint) ----------------
typedef __attribute__((address_space(3))) const void lds_cv;
static __device__ __forceinline__ unsigned int lds_off(const void* p) {
    return (unsigned int)(unsigned long long)(lds_cv*)p;
}

#if defined(__has_builtin)
#if __has_builtin(__builtin_amdgcn_tensor_load_to_lds)
#define HAVE_TDM 1
#endif
#endif

#ifdef HAVE_TDM
// Build D# (ISA ch.8) and issue TENSOR_LOAD_TO_LDS (6-arg clang-23 form).
// Loads tile [nrel][rows][KPAD] f32 from gbase (dim0=KPAD contiguous,
// dim1 row stride = KPAD, dim2 relation stride = rel_stride elements).
static __device__ __forceinline__ void tdm_load_tile(const float* gbase,
                                                     unsigned int lds_byte_off,
                                                     unsigned int rows,
                                                     unsigned int tensor_rows,
                                                     unsigned int nrel,
                                                     unsigned int rel_stride) {
    unsigned long long ga = (unsigned long long)gbase;
    v4u g0;
    g0.x = 1u;                                            // count=1, user descriptor
    g0.y = lds_byte_off;                                  // lds_addr [63:32]
    g0.z = (unsigned int)(ga & 0xFFFFFFFFu);              // global_addr [95:64]
    g0.w = (unsigned int)((ga >> 32) & 0x1FFFFFFu)        // global_addr [120:96]
         | (2u << 30);                                    // type=2 ("image") [127:126]
    v8i g1;
    g1[0] = (int)(2u << 16);                              // wg_mask=0, data_size=4B
    g1[1] = (int)((unsigned)KPAD << 16);                  // tensor_dim0[15:0] @ bits63:48
    g1[2] = (int)((tensor_rows & 0xFFFFu) << 16);         // dim0 hi=0, tensor_dim1[15:0]
    g1[3] = (int)((tensor_rows >> 16) | ((unsigned)KPAD << 16)); // dim1 hi, tile_dim0
    g1[4] = (int)(rows | ((nrel > 1u ? nrel : 0u) << 16));       // tile_dim1, tile_dim2
    g1[5] = (int)KPAD;                                    // tensor_dim0_stride lo32
    g1[6] = (int)((rel_stride & 0xFFFFu) << 16);          // d0_stride hi=0, d1_stride[15:0]
    g1[7] = (int)(rel_stride >> 16);                      // tensor_dim1_stride[47:16]
    v4i g2 = { (int)(nrel > 1u ? nrel : 1u), 0, 0, 0 };   // tensor_dim2
    v4i g3 = { 0, 0, 0, 0 };
    v8i g4 = { 0, 0, 0, 0, 0, 0, 0, 0 };                  // unused extra group (zeros)
    __builtin_amdgcn_tensor_load_to_lds(g0, g1, g2, g3, g4, 0);
}
#endif

// ---------------- kernel 0: zero scratch ----------------
__global__ void GAE_zero(float* __restrict__ p, size_t n) {
    size_t stride = (size_t)gridDim.x * blockDim.x;
    for (size_t i = (size_t)blockIdx.x * blockDim.x + threadIdx.x; i < n; i += stride)
        p[i] = 0.0f;
}

// ---------------- kernel 1: weight = relu(cumsum(ord_basis)) ----------------
__global__ void GAE_cumsum_relu(const float* __restrict__ ob, float* __restrict__ w) {
    const size_t per = (size_t)N_NODES * HID;
    size_t i = (size_t)blockIdx.x * blockDim.x + threadIdx.x;
    if (i >= per) return;
    float s = 0.0f;
#pragma unroll
    for (int r = 0; r < NREL; ++r) {
        s += ob[(size_t)r * per + i];
        w[(size_t)r * per + i] = fmaxf(s, 0.0f);
    }
}

// ---------------- kernel 2: edge gather * norm -> scatter-add ----------------
__global__ void GAE_scatter(const int* __restrict__ x, const int* __restrict__ ei,
                            const int* __restrict__ et, const float* __restrict__ en,
                            const float* __restrict__ w, float* __restrict__ feats) {
    const int e = blockIdx.x;
    const int src = ei[e];
    const int dst = ei[NEDGE + e];
    const int t   = et[e];
    const float nrm = en[e];
    const float* __restrict__ wrow = w + ((size_t)t * N_NODES + x[src]) * HID;
    float* __restrict__ frow = feats + (size_t)dst * HID;
    for (int c = threadIdx.x; c < HID; c += blockDim.x)
        atomicAdd(&frow[c], wrow[c] * nrm);
}

// ---------------- kernel 3: proj = feats @ fc_w.T  (WMMA f32 16x16x4) ----------------
__global__ void GAE_proj_wmma(const float* __restrict__ feats, const float* __restrict__ fcw,
                              float* __restrict__ pu, float* __restrict__ pi) {
    const int lane = threadIdx.x & 31;
    const int half = lane >> 4;
    const int lm   = lane & 15;
    const int m0 = blockIdx.y * 16;
    const int n0 = blockIdx.x * 16;
    const int am = min(m0 + lm, N_NODES - 1);
    const int bn = min(n0 + lm, OUTC - 1);
    const float* __restrict__ arow = feats + (size_t)am * HID;
    const float* __restrict__ brow = fcw   + (size_t)bn * HID;   // fc_w[n][k] == fc_w.T[k][n]
    v8f acc = {};
    for (int k = 0; k < HID; k += 4) {
        const int kk = k + 2 * half;
        v2f a, b;
        a.x = arow[kk]; a.y = arow[kk + 1];
        b.x = brow[kk]; b.y = brow[kk + 1];
        acc = __builtin_amdgcn_wmma_f32_16x16x4_f32(false, a, false, b, (short)0, acc,
                                                    false, false);
    }
    const int n = n0 + lm;
    if (n < OUTC) {
#pragma unroll
        for (int j = 0; j < 8; ++j) {
            const int row = m0 + j + 8 * half;
            if (row < N_USER)        pu[(size_t)row * KPAD + n] = acc[j];
            else if (row < N_NODES)  pi[(size_t)(row - N_USER) * KPAD + n] = acc[j];
        }
    }
}

// ---------------- kernel 4: Q[r] = coefs[r] @ basis_matrix ----------------
__global__ void GAE_qmat(const float* __restrict__ coefs, const float* __restrict__ bm,
                         float* __restrict__ q) {
    const int tot = NREL * OUTC * OUTC;
    int i = blockIdx.x * blockDim.x + threadIdx.x;
    if (i >= tot) return;
    const int r  = i / (OUTC * OUTC);
    const int fg = i % (OUTC * OUTC);
    float s = 0.0f;
#pragma unroll
    for (int b = 0; b < NBASIS; ++b)
        s += coefs[r * NBASIS + b] * bm[(size_t)b * (OUTC * OUTC) + fg];
    q[i] = s;
}

// ---------------- kernel 5: uQ[r] = projU @ Q[r] ----------------
__global__ void GAE_uq(const float* __restrict__ pu, const float* __restrict__ q,
                       float* __restrict__ uq) {
    const int tot = NREL * UPAD * OUTC;
    int i = blockIdx.x * blockDim.x + threadIdx.x;
    if (i >= tot) return;
    const int g = i % OUTC;
    const int t = i / OUTC;
    const int u = t % UPAD;
    const int r = t / UPAD;
    const float* __restrict__ pr = pu + (size_t)u * KPAD;
    const float* __restrict__ qr = q + (size_t)r * OUTC * OUTC + g;
    float s = 0.0f;
    for (int f = 0; f < OUTC; ++f) s += pr[f] * qr[(size_t)f * OUTC];
    uq[((size_t)r * UPAD + u) * KPAD + g] = s;
}

// ---------------- kernel 6: fused bilinear scores + log_softmax ----------------
// 256 threads = 8 waves per block; block computes a 32u x 64i macro-tile.
// TDM stages uQ slab [5][32][80] + projI slab [64][80] into LDS (2 DMA ops),
// then each wave does its 16x16 tile: 20 k-steps x 5 relations of
// v_wmma_f32_16x16x4_f32 with fragments from ds_load.
__global__ void GAE_scores_wmma(const float* __restrict__ uq, const float* __restrict__ pi,
                                float* __restrict__ out) {
    __shared__ float smem[SA_F + SB_F];
    const int tid  = threadIdx.x;
    const int lane = tid & 31;
    const int wvid = tid >> 5;          // 0..7
    const int half = lane >> 4;
    const int lm   = lane & 15;
    const int usub = wvid >> 2;         // 0..1
    const int isub = wvid & 3;          // 0..3
    const int u0 = blockIdx.y * BU;     // 0..6016
    const int i0 = blockIdx.x * BI;     // 0..3648

#ifdef HAVE_TDM
    if (wvid == 0) {
        // uQ slab: 3D tile (K=80, rows=32, rel=5), relation stride UPAD*KPAD
        tdm_load_tile(uq + (size_t)u0 * KPAD, lds_off(smem),
                      BU, UPAD, NREL, (unsigned)(UPAD * KPAD));
        // projI slab: 2D tile (K=80, rows=64)
        tdm_load_tile(pi + (size_t)i0 * KPAD, lds_off(smem + SA_F),
                      BI, IPAD, 1u, 0u);
        __builtin_amdgcn_s_wait_tensorcnt(0);
    }
#else
    for (int t4 = tid; t4 < (SA_F >> 2); t4 += 256) {
        const int e   = t4 << 2;
        const int rel = e / (BU * KPAD);
        const int rem = e - rel * (BU * KPAD);
        const int row = rem / KPAD, k = rem - row * KPAD;
        *(float4*)(smem + e) =
            *(const float4*)(uq + ((size_t)rel * UPAD + u0 + row) * KPAD + k);
    }
    for (int t4 = tid; t4 < (SB_F >> 2); t4 += 256) {
        const int e = t4 << 2;
        const int row = e / KPAD, k = e - row * KPAD;
        *(float4*)(smem + SA_F + e) =
            *(const float4*)(pi + ((size_t)(i0 + row)) * KPAD + k);
    }
#endif
    __syncthreads();

    const float* __restrict__ brow = smem + SA_F + (isub * 16 + lm) * KPAD;
    const float* __restrict__ arow = smem + (usub * 16 + lm) * KPAD;
    const int rsl = BU * KPAD;          // 2560 floats per relation in LDS

    v8f a0 = {}, a1 = {}, a2 = {}, a3 = {}, a4 = {};
#pragma unroll 4
    for (int k = 0; k < KPAD; k += 4) {
        const int kk = k + 2 * half;
        v2f b; b.x = brow[kk]; b.y = brow[kk + 1];     // shared across 5 relations
        v2f a;
        a.x = arow[kk];           a.y = arow[kk + 1];
        a0 = __builtin_amdgcn_wmma_f32_16x16x4_f32(false, a, false, b, (short)0, a0, false, false);
        a.x = arow[rsl + kk];     a.y = arow[rsl + kk + 1];
        a1 = __builtin_amdgcn_wmma_f32_16x16x4_f32(false, a, false, b, (short)0, a1, false, false);
        a.x = arow[2*rsl + kk];   a.y = arow[2*rsl + kk + 1];
        a2 = __builtin_amdgcn_wmma_f32_16x16x4_f32(false, a, false, b, (short)0, a2, false, false);
        a.x = arow[3*rsl + kk];   a.y = arow[3*rsl + kk + 1];
        a3 = __builtin_amdgcn_wmma_f32_16x16x4_f32(false, a, false, b, (short)0, a3, false, false);
        a.x = arow[4*rsl + kk];   a.y = arow[4*rsl + kk + 1];
        a4 = __builtin_amdgcn_wmma_f32_16x16x4_f32(false, a, false, b, (short)0, a4, false, false);
    }

    const int ni = i0 + isub * 16 + lm;
    if (ni >= N_ITEM) return;
#pragma unroll
    for (int j = 0; j < 8; ++j) {
        const int nu = u0 + usub * 16 + j + 8 * half;
        if (nu >= N_USER) continue;
        float s0 = a0[j], s1 = a1[j], s2 = a2[j], s3 = a3[j], s4 = a4[j];
        float m = fmaxf(fmaxf(fmaxf(s0, s1), fmaxf(s2, s3)), s4);
        float sum = expf(s0 - m) + expf(s1 - m) + expf(s2 - m) + expf(s3 - m) + expf(s4 - m);
        float lse = m + logf(sum);
        float* __restrict__ o = out + ((size_t)nu * N_ITEM + ni) * NREL;
        o[0] = s0 - lse; o[1] = s1 - lse; o[2] = s2 - lse; o[3] = s3 - lse; o[4] = s4 - lse;
    }
}

extern "C" void kernel_launch(void* const* d_in, const int* in_sizes, int n_in,
                              void* d_out, int out_size, void* d_ws, size_t ws_size,
                              hipStream_t stream) {
    (void)in_sizes; (void)n_in; (void)out_size; (void)ws_size;
    const int*   x     = (const int*)d_in[0];
    const int*   ei    = (const int*)d_in[1];
    const int*   et    = (const int*)d_in[2];
    const float* en    = (const float*)d_in[3];
    const float* ob    = (const float*)d_in[4];
    const float* fcw   = (const float*)d_in[5];
    const float* bm    = (const float*)d_in[6];
    const float* coefs = (const float*)d_in[7];
    float* out = (float*)d_out;
    float* ws  = (float*)d_ws;

    float* w     = ws + OFF_W;
    float* feats = ws + OFF_F;
    float* pu    = ws + OFF_PU;
    float* pi    = ws + OFF_PI;
    float* q     = ws + OFF_Q;
    float* uq    = ws + OFF_UQ;

    GAE_zero<<<2048, 256, 0, stream>>>(feats, ZERO_N);

    {
        const size_t per = (size_t)N_NODES * HID;
        GAE_cumsum_relu<<<(int)((per + 255) / 256), 256, 0, stream>>>(ob, w);
    }

    GAE_scatter<<<NEDGE, 128, 0, stream>>>(x, ei, et, en, w, feats);

    GAE_proj_wmma<<<dim3((OUTC + 15) / 16, (N_NODES + 15) / 16), 32, 0, stream>>>(
        feats, fcw, pu, pi);

    GAE_qmat<<<(NREL * OUTC * OUTC + 255) / 256, 256, 0, stream>>>(coefs, bm, q);

    GAE_uq<<<(NREL * UPAD * OUTC + 255) / 256, 256, 0, stream>>>(pu, q, uq);

    // 58 i-macro-tiles x 189 u-macro-tiles, 8 waves/block
    GAE_scores_wmma<<<dim3(IPAD / BI, UPAD / BU), 256, 0, stream>>>(uq, pi, out);
}